// VIN_26903675142386
// MI455X (gfx1250) — compile-verified
//
#include <hip/hip_runtime.h>

typedef __attribute__((ext_vector_type(2))) float v2f;
typedef __attribute__((ext_vector_type(8))) float v8f;

#define VSTR 68                       // padded row stride
#define PROWS 66                      // 64 + 2 pad rows
#define QRT_FLOATS (256 * 256)        // qrT[256 tiles][32 lanes][8] in WMMA C layout
#define RPAD_FLOATS (PROWS * VSTR)    // 4488
#define RPAD_OFF  QRT_FLOATS
#define VPAD_OFF  (RPAD_OFF + RPAD_FLOATS)
#define WC_OFF    (VPAD_OFF + 2 * RPAD_FLOATS)
#define LDS_FLOATS (WC_OFF + 32)      // 79032 floats = 316128 B <= 320 KB/WGP
#define NEG_BIG (-3.0e38f)

// Tap -> K-slot table, chosen so each WMMA's two B values share one per-lane
// base address with identical immediate offsets in both half-waves
// (enables ds_load_2addr_b32 merging).  -1 = zero slot.
//   wmma0: K0..K3 = taps (0,0),(0,1) | (1,0),(1,1)
//   wmma1: K0..K3 = taps (0,2),(1,2) | (2,2),zero
//   wmma2: K0..K3 = taps (2,0),(2,1) | zero,zero
__device__ __forceinline__ void build_A(const float* __restrict__ Wg, int lane, v2f* a) {
  const int T[3][4] = {{0, 1, 3, 4}, {2, 5, 8, -1}, {6, 7, -1, -1}};
  const int M = lane & 15, half = lane >> 4;
#pragma unroll
  for (int i = 0; i < 3; ++i) {
    const int ta = T[i][2 * half], tb = T[i][2 * half + 1];
    a[i].x = (M < 10 && ta >= 0) ? Wg[M * 9 + ta] : 0.f;
    a[i].y = (M < 10 && tb >= 0) ? Wg[M * 9 + tb] : 0.f;
  }
}

// D = A(10x9 filters, padded 16x12) * im2col(src 3x3 taps, 16 positions) + C.
// B layout mirrors C: VGPR0 = K row (lanes 0-15) / K+2 row (lanes 16-31).
__device__ __forceinline__ v8f conv3x3_wmma(v2f a0, v2f a1, v2f a2,
                                            const float* __restrict__ src,
                                            int y, int x, int half, v8f c) {
  const float* p = src + y * VSTR + x;
  v2f b;
  { // wmma0: taps (0,0),(0,1) | (1,0),(1,1)  -> read2 offsets {0,1}
    const float* b0 = p + (half ? VSTR : 0);
    b.x = b0[0];
    b.y = b0[1];
    c = __builtin_amdgcn_wmma_f32_16x16x4_f32(false, a0, false, b, (short)0, c, false, false);
  }
  { // wmma1: taps (0,2),(1,2) | (2,2),zero   -> read2 offsets {0,VSTR}
    const float* b1 = p + 2 + (half ? 2 * VSTR : 0);
    const float t0 = b1[0], t1 = b1[VSTR];
    b.x = t0;
    b.y = half ? 0.f : t1;          // h1's second slot is a zero pad
    c = __builtin_amdgcn_wmma_f32_16x16x4_f32(false, a1, false, b, (short)0, c, false, false);
  }
  { // wmma2: taps (2,0),(2,1) | zero,zero    -> read2 offsets {0,1}
    const float* b2 = p + 2 * VSTR;
    const float u0 = b2[0], u1 = b2[1];
    b.x = half ? 0.f : u0;
    b.y = half ? 0.f : u1;
    c = __builtin_amdgcn_wmma_f32_16x16x4_f32(false, a2, false, b, (short)0, c, false, false);
  }
  return c;
}

__global__ __launch_bounds__(256) void vin_kernel(
    const float* __restrict__ input,   // (128,2,64,64)
    const int*   __restrict__ sx,      // (128)
    const int*   __restrict__ sy,      // (128)
    const int*   __restrict__ kptr,    // scalar k
    const float* __restrict__ hw,      // (150,2,3,3)
    const float* __restrict__ hb,      // (150)
    const float* __restrict__ rw,      // (150) (1,150,1,1)
    const float* __restrict__ qw,      // (10,9)
    const float* __restrict__ ww,      // (10,9)
    const float* __restrict__ fcw,     // (8,10)
    float* __restrict__ out)           // logits[128,8] ++ softmax[128,8]
{
  extern __shared__ float smem[];
  float* qrT  = smem;              // WMMA-C-layout qr tiles (input staged here pre-qr)
  float* rpad = smem + RPAD_OFF;   // 66x68 padded r
  float* vpad = smem + VPAD_OFF;   // 2 x 66x68 double-buffered v
  float* wc   = smem + WC_OFF;     // collapsed h->r effective weights [18] + bias [1]

  const int tid  = threadIdx.x;
  const int b    = blockIdx.x;
  const int lane = tid & 31;
  const int wave = tid >> 5;
  const int half = (lane >> 4) & 1;
  const int n16  = lane & 15;

  // ---- P1: stage input into LDS; collapse (h_w,h_b,r_w) -> 2x3x3 conv; zero pads
  {
    const float4* i4 = (const float4*)(input + b * (2 * 4096));
    float4* u4 = (float4*)smem;
    for (int i = tid; i < 2048; i += 256) u4[i] = i4[i];
    if (tid < 19) {
      float s = 0.f;
      if (tid < 18) { for (int c = 0; c < 150; ++c) s += rw[c] * hw[c * 18 + tid]; }
      else          { for (int c = 0; c < 150; ++c) s += rw[c] * hb[c]; }
      wc[tid] = s;
    }
    for (int i = tid; i < RPAD_FLOATS;     i += 256) rpad[i] = 0.f;
    for (int i = tid; i < 2 * RPAD_FLOATS; i += 256) vpad[i] = 0.f;
  }
  __syncthreads();

  // ---- P2: r = conv3x3(input, r_eff) + r_eff_b into padded plane
  {
    const float reffb = wc[18];
    for (int p = tid; p < 4096; p += 256) {
      const int y = p >> 6, x = p & 63;
      float acc = reffb;
#pragma unroll
      for (int i = 0; i < 2; ++i)
#pragma unroll
        for (int t = 0; t < 9; ++t) {
          const int yy = y + t / 3 - 1, xx = x + t % 3 - 1;
          if (yy >= 0 && yy < 64 && xx >= 0 && xx < 64)
            acc += wc[i * 9 + t] * smem[i * 4096 + (yy << 6) + xx];
        }
      rpad[(y + 1) * VSTR + x + 1] = acc;
    }
  }
  __syncthreads();

  // ---- P3: qr = conv3x3(r, q_w) via WMMA, stored in C-register tile layout
  //          (channels 10..15 pre-filled with -BIG); v0 = max_ch(qr).
  {
    v2f aq[3];
    build_A(qw, lane, aq);
    for (int t = 0; t < 32; ++t) {
      const int y = (wave << 3) + (t >> 2);
      const int x = ((t & 3) << 4) + n16;
      const int tile = (y << 2) + (t & 3);
      v8f c = {};
      v8f d = conv3x3_wmma(aq[0], aq[1], aq[2], rpad, y, x, half, c);
      v8f s;
      float m = NEG_BIG;
#pragma unroll
      for (int j = 0; j < 8; ++j) {
        const bool valid = (half == 0) || (j < 2);   // channel j + 8*half < 10
        s[j] = valid ? d[j] : NEG_BIG;
        m = fmaxf(m, s[j]);
      }
      *(v8f*)(qrT + tile * 256 + lane * 8) = s;      // 2x ds_store_b128
      m = fmaxf(m, __shfl_xor(m, 16, 32));
      vpad[(y + 1) * VSTR + x + 1] = m;              // both halves write same value
    }
  }
  __syncthreads();

  // ---- P4: k-1 value-iteration steps, fully LDS-resident.
  //          D = w * im2col(v) + qrT(tile) ; v' = max over channels of D
  //          (pad channels carry -BIG from the C operand, so a plain max works).
  const int kk = kptr[0];
  {
    v2f aw3[3];
    build_A(ww, lane, aw3);
    for (int it = 0; it < kk - 1; ++it) {
      const float* vsrc = vpad + (it & 1) * RPAD_FLOATS;
      float*       vdst = vpad + ((it & 1) ^ 1) * RPAD_FLOATS;
      for (int t = 0; t < 32; ++t) {
        const int y = (wave << 3) + (t >> 2);
        const int x = ((t & 3) << 4) + n16;
        const int tile = (y << 2) + (t & 3);
        v8f c = *(const v8f*)(qrT + tile * 256 + lane * 8);  // 2x ds_load_b128
        v8f d = conv3x3_wmma(aw3[0], aw3[1], aw3[2], vsrc, y, x, half, c);
        float m = d[0];
#pragma unroll
        for (int j = 1; j < 8; ++j) m = fmaxf(m, d[j]);
        m = fmaxf(m, __shfl_xor(m, 16, 32));
        vdst[(y + 1) * VSTR + x + 1] = m;
      }
      __syncthreads();
    }
  }

  // ---- P5: final q at (sx,sy), FC, softmax
  if (tid == 0) {
    const float* vfin = vpad + ((kk - 1) & 1) * RPAD_FLOATS;
    const int xs = sx[b], ys = sy[b];
    const int tile = (xs << 2) + (ys >> 4);
    const int li = ys & 15;
    float qo[10];
#pragma unroll
    for (int a = 0; a < 10; ++a) {
      const int ln = li + ((a & 8) ? 16 : 0);
      float acc = qrT[tile * 256 + ln * 8 + (a & 7)];
#pragma unroll
      for (int t = 0; t < 9; ++t)
        acc += ww[a * 9 + t] * vfin[(xs + t / 3) * VSTR + (ys + t % 3)];
      qo[a] = acc;
    }
    float lg[8], mx = NEG_BIG;
#pragma unroll
    for (int o = 0; o < 8; ++o) {
      float s = 0.f;
#pragma unroll
      for (int a = 0; a < 10; ++a) s += fcw[o * 10 + a] * qo[a];
      lg[o] = s; mx = fmaxf(mx, s);
    }
    float den = 0.f, ex[8];
#pragma unroll
    for (int o = 0; o < 8; ++o) { ex[o] = expf(lg[o] - mx); den += ex[o]; }
#pragma unroll
    for (int o = 0; o < 8; ++o) {
      out[b * 8 + o]        = lg[o];
      out[1024 + b * 8 + o] = ex[o] / den;
    }
  }
}

extern "C" void kernel_launch(void* const* d_in, const int* in_sizes, int n_in,
                              void* d_out, int out_size, void* d_ws, size_t ws_size,
                              hipStream_t stream) {
  (void)in_sizes; (void)n_in; (void)out_size; (void)d_ws; (void)ws_size;
  const size_t shmem = LDS_FLOATS * sizeof(float);   // ~316 KB dynamic LDS (<= 320 KB/WGP)
  vin_kernel<<<128, 256, shmem, stream>>>(
      (const float*)d_in[0], (const int*)d_in[1], (const int*)d_in[2],
      (const int*)d_in[3],
      (const float*)d_in[4], (const float*)d_in[5], (const float*)d_in[6],
      (const float*)d_in[7], (const float*)d_in[8], (const float*)d_in[9],
      (float*)d_out);
}